// GraphConvLayer_50818053046787
// MI455X (gfx1250) — compile-verified
//
#include <hip/hip_runtime.h>
#include <hip/hip_bf16.h>

typedef __attribute__((ext_vector_type(2))) float v2f;
typedef __attribute__((ext_vector_type(8))) float v8f;

#define N_NODES 50000
#define N_EDGES 800000
#define DIM 64

// ---------------------------------------------------------------------------
// Zero scratch (agg + degree). Deterministic re-init every launch.
// ---------------------------------------------------------------------------
__global__ void gc_zero_kernel(float* __restrict__ p, int n) {
    int i = blockIdx.x * blockDim.x + threadIdx.x;
    if (i < n) p[i] = 0.0f;
}

// ---------------------------------------------------------------------------
// X = A @ W^T + b_lin  using V_WMMA_F32_16X16X4_F32 (exact fp32).
// One wave computes one 16x16 output tile; block = 4 waves = the 4 col tiles.
// gridDim.x = N/16 row tiles.
//
// A-operand (16x4 f32, 2 VGPRs): lane<16 -> row M=lane, K = k0+{0,1};
//                                lane>=16 -> row M=lane-16, K = k0+{2,3}.
// B-operand (4x16): B[k,n] = W[n,k]; identical per-lane addressing on W's rows.
// C/D (16x16 f32, 8 VGPRs): VGPR j -> row j (lanes 0-15) / row j+8 (lanes 16-31).
// ---------------------------------------------------------------------------
__global__ __launch_bounds__(128) void gc_gemm_wmma_kernel(
    const float* __restrict__ A,      // [N, 64] node features
    const float* __restrict__ W,      // [64, 64]
    const float* __restrict__ b_lin,  // [64]
    float* __restrict__ X)            // [N, 64] output
{
    const int wave = threadIdx.x >> 5;      // 0..3 -> column tile
    const int lane = threadIdx.x & 31;
    const int rowTile = blockIdx.x;         // 0..N/16-1
    const int m16   = lane & 15;
    const int kHalf = (lane >> 4) << 1;     // 0 or 2

    const float* arow = A + (size_t)(rowTile * 16 + m16) * DIM;
    const float* wrow = W + (size_t)(wave   * 16 + m16) * DIM;

    v8f c = {0.f, 0.f, 0.f, 0.f, 0.f, 0.f, 0.f, 0.f};
#pragma unroll
    for (int k0 = 0; k0 < DIM; k0 += 4) {
        v2f a = *(const v2f*)(arow + k0 + kHalf);
        v2f b = *(const v2f*)(wrow + k0 + kHalf);
        // (neg_a, A, neg_b, B, c_mod, C, reuse_a, reuse_b)
        c = __builtin_amdgcn_wmma_f32_16x16x4_f32(
                false, a, false, b, (short)0, c, false, false);
    }

    const int col   = wave * 16 + m16;
    const float bl  = b_lin[col];
    const int rbase = rowTile * 16 + ((lane >> 4) << 3);  // +8 for upper half-wave
#pragma unroll
    for (int j = 0; j < 8; ++j) {
        X[(size_t)(rbase + j) * DIM + col] = c[j] + bl;
    }
}

// ---------------------------------------------------------------------------
// degree[dst] += 1 per edge
// ---------------------------------------------------------------------------
__global__ void gc_degree_kernel(const int* __restrict__ ei, float* __restrict__ deg, int E) {
    int e = blockIdx.x * blockDim.x + threadIdx.x;
    if (e < E) atomicAdd(&deg[ei[2 * e + 1]], 1.0f);
}

// ---------------------------------------------------------------------------
// agg[dst] += X[src] : one wave per edge, each lane handles a float2 (64 feats).
// Gather of X[src] is a coalesced 256B wave read (stays in L2: X is 12.8MB).
// ---------------------------------------------------------------------------
__global__ void gc_scatter_kernel(const int* __restrict__ ei,
                                  const float* __restrict__ X,
                                  float* __restrict__ agg, int E) {
    int tid  = blockIdx.x * blockDim.x + threadIdx.x;
    int e    = tid >> 5;
    if (e >= E) return;
    int lane = tid & 31;
    int src  = ei[2 * e];
    int dst  = ei[2 * e + 1];
    const float2 v = *(const float2*)(X + (size_t)src * DIM + lane * 2);
    float* dp = agg + (size_t)dst * DIM + lane * 2;
    atomicAdd(dp + 0, v.x);
    atomicAdd(dp + 1, v.y);
}

// ---------------------------------------------------------------------------
// out = relu(X + agg/max(deg,1) + bias)
// ---------------------------------------------------------------------------
__global__ void gc_finalize_kernel(const float* __restrict__ X,
                                   const float* __restrict__ agg,
                                   const float* __restrict__ deg,
                                   const float* __restrict__ bias,
                                   float* __restrict__ out, int total) {
    int i = blockIdx.x * blockDim.x + threadIdx.x;
    if (i >= total) return;
    int n = i >> 6;
    int f = i & 63;
    float d = fmaxf(deg[n], 1.0f);
    float v = X[i] + agg[i] / d + bias[f];
    out[i] = fmaxf(v, 0.0f);
}

extern "C" void kernel_launch(void* const* d_in, const int* in_sizes, int n_in,
                              void* d_out, int out_size, void* d_ws, size_t ws_size,
                              hipStream_t stream) {
    const float* node_features = (const float*)d_in[0];  // [N,64] f32
    const int*   edge_index    = (const int*)d_in[1];    // [E,2]  i32 (JAX x64 off)
    const float* W             = (const float*)d_in[2];  // [64,64] f32
    const float* b_lin         = (const float*)d_in[3];  // [64]
    const float* bias          = (const float*)d_in[4];  // [64]
    float*       out           = (float*)d_out;          // [N,64]

    // Workspace layout: X [N*64] | agg [N*64] | degree [N]
    float* X   = (float*)d_ws;
    float* agg = X + (size_t)N_NODES * DIM;
    float* deg = agg + (size_t)N_NODES * DIM;

    // 1) zero agg + degree (contiguous region)
    {
        int n = N_NODES * DIM + N_NODES;
        gc_zero_kernel<<<(n + 255) / 256, 256, 0, stream>>>(agg, n);
    }

    // 2) GEMM via f32 WMMA: N/16 row tiles, 4 col-tile waves per block
    gc_gemm_wmma_kernel<<<N_NODES / 16, 128, 0, stream>>>(node_features, W, b_lin, X);

    // 3) degree
    gc_degree_kernel<<<(N_EDGES + 255) / 256, 256, 0, stream>>>(edge_index, deg, N_EDGES);

    // 4) scatter-add: one wave (32 lanes) per edge
    {
        long long threads = (long long)N_EDGES * 32;
        gc_scatter_kernel<<<(int)((threads + 255) / 256), 256, 0, stream>>>(edge_index, X, agg, N_EDGES);
    }

    // 5) finalize
    {
        int total = N_NODES * DIM;
        gc_finalize_kernel<<<(total + 255) / 256, 256, 0, stream>>>(X, agg, deg, bias, out, total);
    }
}